// SpatialAsymmetryDetector_1451698946808
// MI455X (gfx1250) — compile-verified
//
#include <hip/hip_runtime.h>

typedef float v2f __attribute__((ext_vector_type(2)));
typedef float v8f __attribute__((ext_vector_type(8)));

#define B_  8
#define T_  4
#define C_  512
#define H_  32
#define W_  32
#define HW_ 1024
#define HP_ 28
#define EPS 1e-5f

// Banded ones matrix for the separable 5x5 VALID box filter:
// S[i][j] = 1 iff output col j is valid (j < 28) and j <= i <= j+4.
__device__ __forceinline__ float sband(int i, int j) {
  return (j < HP_ && i >= j && i <= j + 4) ? 1.0f : 0.0f;
}

// ---------------------------------------------------------------------------
// Kernel 1: per-(t,c) BN statistics over (B,H,W)=8192 elems for both tensors,
// fused into 3 coefficients:  diff_c = A*xl - B*xr + C   (beta cancels).
// One pass over all 128 MiB, fully coalesced; leaves data resident in L2.
// ---------------------------------------------------------------------------
__global__ __launch_bounds__(256) void sad_stats_kernel(
    const float* __restrict__ L, const float* __restrict__ R,
    const float* __restrict__ gamma, float* __restrict__ coef /* [3][T_*C_] */) {
  const int tc  = blockIdx.x;            // t*C_ + c
  const int t   = tc / C_;
  const int c   = tc % C_;
  const int tid = threadIdx.x;

  float sl = 0.f, sl2 = 0.f, sr = 0.f, sr2 = 0.f;
  for (int e = tid; e < B_ * HW_; e += 256) {
    const int b  = e >> 10;
    const int hw = e & (HW_ - 1);
    const size_t addr = ((((size_t)b * T_ + t) * C_ + c) * HW_) + hw;
    const float xl = L[addr];
    const float xr = R[addr];
    sl += xl; sl2 += xl * xl;
    sr += xr; sr2 += xr * xr;
  }

  __shared__ float s0[256], s1[256], s2[256], s3[256];
  s0[tid] = sl; s1[tid] = sl2; s2[tid] = sr; s3[tid] = sr2;
  __syncthreads();
  for (int off = 128; off > 0; off >>= 1) {
    if (tid < off) {
      s0[tid] += s0[tid + off];
      s1[tid] += s1[tid + off];
      s2[tid] += s2[tid + off];
      s3[tid] += s3[tid + off];
    }
    __syncthreads();
  }

  if (tid == 0) {
    const float invn = 1.0f / (float)(B_ * HW_);
    const float mul  = s0[0] * invn;
    const float varl = s1[0] * invn - mul * mul;
    const float mur  = s2[0] * invn;
    const float varr = s3[0] * invn - mur * mur;
    const float g    = gamma[c];
    const float A    = g / sqrtf(varl + EPS);
    const float Bc   = g / sqrtf(varr + EPS);
    coef[tc]                 = A;
    coef[T_ * C_ + tc]       = Bc;
    coef[2 * T_ * C_ + tc]   = Bc * mur - A * mul;
  }
}

// ---------------------------------------------------------------------------
// Kernel 2 (fused): per-(b,t) block of 1024 threads.
//  Stage A: channel reduction  diff2[h,w] = sum_c (A*xl - B*xr + C)^2
//           (one coalesced pass, expected to hit the 192 MB L2)
//  Stage B: wave 0 does the 5x5 box filter as two banded matmuls with
//           V_WMMA_F32_16X16X4_F32, then sqrt, max, argmin, and writes
//           all outputs. Output layout (floats, concatenated):
//           [0,32): asymmetry_values  [32,96): coords (x,y)  [96,...): heatmap
// ---------------------------------------------------------------------------
__global__ __launch_bounds__(1024) void sad_heatmap_kernel(
    const float* __restrict__ L, const float* __restrict__ R,
    const float* __restrict__ coef, float* __restrict__ out) {
  const int bt  = blockIdx.x;            // b*T_ + t
  const int t   = bt % T_;
  const int tid = threadIdx.x;

  __shared__ float cA[C_], cB[C_], cC[C_];
  __shared__ float sD[H_ * 33];          // diff2 tile, stride 33 (bank-safe)
  __shared__ float sW[H_ * 33];          // horizontal window sums
  __shared__ float rmax[32], rmin[32];
  __shared__ int   ridx[32];

  if (tid < C_) {
    cA[tid] = coef[t * C_ + tid];
    cB[tid] = coef[T_ * C_ + t * C_ + tid];
    cC[tid] = coef[2 * T_ * C_ + t * C_ + tid];
  }
  __syncthreads();

  // ---- Stage A: each thread owns one (h,w); loop over 512 channels ----
  {
    float acc = 0.f;
    const size_t base = (size_t)bt * C_ * HW_ + tid;
    #pragma unroll 4
    for (int c = 0; c < C_; ++c) {
      const float d = fmaf(cA[c], L[base + (size_t)c * HW_],
                      fmaf(-cB[c], R[base + (size_t)c * HW_], cC[c]));
      acc = fmaf(d, d, acc);
    }
    sD[(tid >> 5) * 33 + (tid & 31)] = acc;
  }
  __syncthreads();

  // ---- Stage B: single wave, full EXEC (WMMA requirement) ----
  if (tid < 32) {
    const int lane = tid;
    const int ln   = lane & 15;          // M (A) / N (B,C,D) index
    const int g2   = (lane >> 4) * 2;    // K sub-offset: lanes 16-31 hold K+2
    const int rofs = (g2 << 2);          // row offset 8*(lane/16) for C/D

    // Pass 1: W1 = D(32x32) * S(32x32banded)  -> horizontal 1x5 sums
    v8f acc1[2][2];
    #pragma unroll
    for (int mi = 0; mi < 2; ++mi)
      #pragma unroll
      for (int ni = 0; ni < 2; ++ni) {
        v8f a_c = {0.f, 0.f, 0.f, 0.f, 0.f, 0.f, 0.f, 0.f};
        #pragma unroll
        for (int kk = 0; kk < 8; ++kk) {
          const int kb = kk * 4 + g2;
          v2f av, bv;
          av.x = sD[(mi * 16 + ln) * 33 + kb];
          av.y = sD[(mi * 16 + ln) * 33 + kb + 1];
          const int ncol = ni * 16 + ln;
          bv.x = sband(kb,     ncol);
          bv.y = sband(kb + 1, ncol);
          a_c = __builtin_amdgcn_wmma_f32_16x16x4_f32(
                    false, av, false, bv, (short)0, a_c, false, false);
        }
        acc1[mi][ni] = a_c;
      }
    #pragma unroll
    for (int mi = 0; mi < 2; ++mi)
      #pragma unroll
      for (int ni = 0; ni < 2; ++ni)
        #pragma unroll
        for (int v = 0; v < 8; ++v)
          sW[(mi * 16 + v + rofs) * 33 + ni * 16 + ln] = acc1[mi][ni][v];

    // Pass 2: win = S^T(32x32banded) * W1  -> vertical 5x1 sums
    v8f acc2[2][2];
    #pragma unroll
    for (int mi = 0; mi < 2; ++mi)
      #pragma unroll
      for (int ni = 0; ni < 2; ++ni) {
        v8f a_c = {0.f, 0.f, 0.f, 0.f, 0.f, 0.f, 0.f, 0.f};
        #pragma unroll
        for (int kk = 0; kk < 8; ++kk) {
          const int kb = kk * 4 + g2;
          v2f av, bv;
          av.x = sband(kb,     mi * 16 + ln);   // (S^T)[m][k] = S[k][m]
          av.y = sband(kb + 1, mi * 16 + ln);
          bv.x = sW[kb * 33 + ni * 16 + ln];
          bv.y = sW[(kb + 1) * 33 + ni * 16 + ln];
          a_c = __builtin_amdgcn_wmma_f32_16x16x4_f32(
                    false, av, false, bv, (short)0, a_c, false, false);
        }
        acc2[mi][ni] = a_c;
      }

    // Heatmap + per-(b,t) max / first-argmin
    float lmax = -3.402823466e+38f;
    float lmin =  3.402823466e+38f;
    int   lidx = 0x7fffffff;
    float* hm = out + B_ * T_ * 3 + bt * (HP_ * HP_);
    #pragma unroll
    for (int mi = 0; mi < 2; ++mi)
      #pragma unroll
      for (int ni = 0; ni < 2; ++ni)
        #pragma unroll
        for (int v = 0; v < 8; ++v) {
          const int m = mi * 16 + v + rofs;
          const int n = ni * 16 + ln;
          if (m < HP_ && n < HP_) {
            const float hv = sqrtf(acc2[mi][ni][v] * (1.0f / 25.0f));
            hm[m * HP_ + n] = hv;
            if (hv > lmax) lmax = hv;
            const int fi = m * HP_ + n;
            if (hv < lmin || (hv == lmin && fi < lidx)) { lmin = hv; lidx = fi; }
          }
        }

    rmax[lane] = lmax; rmin[lane] = lmin; ridx[lane] = lidx;
    // single-wave LDS ops are in-order; no barrier needed
    if (lane == 0) {
      float gmax = rmax[0], gmin = rmin[0];
      int   gidx = ridx[0];
      for (int i = 1; i < 32; ++i) {
        if (rmax[i] > gmax) gmax = rmax[i];
        if (rmin[i] < gmin || (rmin[i] == gmin && ridx[i] < gidx)) {
          gmin = rmin[i]; gidx = ridx[i];
        }
      }
      out[bt] = gmax;                                   // asymmetry_values
      out[B_ * T_ + bt * 2 + 0] = (float)(gidx % HP_);  // x_argmin
      out[B_ * T_ + bt * 2 + 1] = (float)(gidx / HP_);  // y_argmin
    }
  }
}

extern "C" void kernel_launch(void* const* d_in, const int* in_sizes, int n_in,
                              void* d_out, int out_size, void* d_ws, size_t ws_size,
                              hipStream_t stream) {
  const float* L     = (const float*)d_in[0];
  const float* R     = (const float*)d_in[1];
  const float* gamma = (const float*)d_in[2];
  // d_in[3] = beta: cancels in (l - r), unused.
  float* coef = (float*)d_ws;     // 3 * T_*C_ = 6144 floats = 24 KiB scratch
  float* out  = (float*)d_out;

  sad_stats_kernel<<<T_ * C_, 256, 0, stream>>>(L, R, gamma, coef);
  sad_heatmap_kernel<<<B_ * T_, 1024, 0, stream>>>(L, R, coef, out);
}